// OptimizedColorMoCAttention_41815801594255
// MI455X (gfx1250) — compile-verified
//
#include <hip/hip_runtime.h>
#include <stdint.h>

// ---------------------------------------------------------------------------
// Fused colorAttention + mean-pool + combine for x:[16,64,256,256] f32.
// One workgroup per (b,c) channel. The whole 256KB channel is staged in LDS
// (CDNA5: 320KB per WGP) via FOUR pipelined 64KB Tensor Data Mover loads:
// wave0 issues all four descriptors, then partial s_wait_tensorcnt(3-c) lets
// waves histogram chunk c while the TDM streams chunks c+1..3 -> DMA latency
// hidden behind compute. x is read from HBM exactly once.
// Memory-bound floor: (256MB in + 256MB out) / 23.3 TB/s ~= 22us.
// ---------------------------------------------------------------------------

typedef unsigned int uint32x4 __attribute__((ext_vector_type(4)));
typedef int          int32x4  __attribute__((ext_vector_type(4)));
typedef int          int32x8  __attribute__((ext_vector_type(8)));

#define HW          65536            // 256*256 pixels per channel
#define TILE_BYTES  (HW * 4)         // 256 KB staged in LDS
#define NCHUNK      4                // pipelined TDM chunks
#define CHUNK_ELEMS (HW / NCHUNK)    // 16384 f32 = 64 KB per chunk
#define NBINS       32
#define NWAVES      32
#define TPB         1024             // 32 wave32 waves
#define F4_PER_CH   (CHUNK_ELEMS / 4)      // 4096 float4 per chunk
#define CH_ITERS    (F4_PER_CH / TPB)      // 4 float4 per thread per chunk

__device__ __forceinline__ void tdm_load_chunk(const float* gsrc, uint32_t lds_byte_off)
{
    const uint64_t ga = (uint64_t)(uintptr_t)gsrc;

    // D# group 0: count=1, lds_addr, 57b global addr, type=2 ("image")
    uint32x4 g0;
    g0[0] = 1u;
    g0[1] = lds_byte_off;
    g0[2] = (uint32_t)ga;
    g0[3] = ((uint32_t)(ga >> 32) & 0x01FFFFFFu) | 0x80000000u;

    // D# group 1: data_size=4B, one row of 16384 contiguous f32 (64 KB)
    int32x8 g1;
    g1[0] = 0x00020000;                         // data_size code 2 (4 bytes)
    g1[1] = (int)((16384u & 0xFFFFu) << 16);    // tensor_dim0[15:0] @ bits 63:48
    g1[2] = (int)((16384u >> 16) | (1u << 16)); // dim0[31:16] | tensor_dim1=1
    g1[3] = (int)(16384u << 16);                // tensor_dim1[31:16]=0 | tile_dim0=16384
    g1[4] = 1;                                  // tile_dim1=1, tile_dim2=0 (unused)
    g1[5] = 16384;                              // tensor_dim0_stride[31:0]
    g1[6] = 0;
    g1[7] = 0;

    int32x4 gz = {0, 0, 0, 0};                  // groups 2/3 unused (<=2D)
#if __clang_major__ >= 23
    int32x8 gz8 = {0, 0, 0, 0, 0, 0, 0, 0};
    __builtin_amdgcn_tensor_load_to_lds(g0, g1, gz, gz, gz8, 0);
#else
    __builtin_amdgcn_tensor_load_to_lds(g0, g1, gz, gz, 0);
#endif
}

// s_wait_tensorcnt needs a literal immediate; switch arms give the frontend
// constants and the unroller folds the switch away.
__device__ __forceinline__ void wait_tensorcnt_le(int n)
{
    switch (n) {
        case 0:  __builtin_amdgcn_s_wait_tensorcnt(0); break;
        case 1:  __builtin_amdgcn_s_wait_tensorcnt(1); break;
        case 2:  __builtin_amdgcn_s_wait_tensorcnt(2); break;
        default: __builtin_amdgcn_s_wait_tensorcnt(3); break;
    }
}

__global__ __launch_bounds__(TPB, 1)
void colormoc_fused_kernel(const float* __restrict__ x, float* __restrict__ out)
{
    extern __shared__ __align__(16) float tile[];          // 256 KB dynamic LDS
    __shared__ unsigned whist[NWAVES * NBINS];             // per-wave histograms
    __shared__ float    ssum;
    __shared__ int      sdom;
    __shared__ float    spooled;

    const int tid  = threadIdx.x;
    const int wave = tid >> 5;
    const int lane = tid & 31;
    const int chan = blockIdx.x;
    const float* xc = x + (size_t)chan * HW;

    // ---- init small LDS state (published by the first chunk barrier) ----
    whist[tid] = 0u;                                       // 1024 entries == TPB
    if (tid == 0) ssum = 0.0f;

    // ---- wave 0 issues all four chunk DMAs back-to-back (TENSORcnt -> 4) ----
    if (tid == 0) {
        const uint32_t lds0 = (uint32_t)(uintptr_t)tile;   // flat LDS low 32 = LDS byte addr
#pragma unroll
        for (int c = 0; c < NCHUNK; ++c)
            tdm_load_chunk(xc + c * CHUNK_ELEMS, lds0 + (uint32_t)c * (CHUNK_ELEMS * 4));
    }

    // ---- pass 1: pipelined histogram + partial sums; chunk c overlaps DMA c+1..3 ----
    const float4* t4 = reinterpret_cast<const float4*>(tile);
    const float kInv = (float)(32.0 / 255.0);              // histc bin scale (f32-rounded)
    unsigned* myh = &whist[wave * NBINS];
    float lsum = 0.0f;

#pragma unroll
    for (int c = 0; c < NCHUNK; ++c) {
        if (tid == 0)
            wait_tensorcnt_le((NCHUNK - 1) - c);           // chunk c landed
        __syncthreads();                                   // publish chunk c to all waves

#pragma unroll
        for (int i = 0; i < CH_ITERS; ++i) {
            float4 v = t4[c * F4_PER_CH + i * TPB + tid];  // ds_load_b128
            {
                int b = (int)floorf(v.x * kInv); b = b < 0 ? 0 : (b > 31 ? 31 : b);
                lsum += v.x; atomicAdd(&myh[b], 1u);       // ds_add_u32
            }
            {
                int b = (int)floorf(v.y * kInv); b = b < 0 ? 0 : (b > 31 ? 31 : b);
                lsum += v.y; atomicAdd(&myh[b], 1u);
            }
            {
                int b = (int)floorf(v.z * kInv); b = b < 0 ? 0 : (b > 31 ? 31 : b);
                lsum += v.z; atomicAdd(&myh[b], 1u);
            }
            {
                int b = (int)floorf(v.w * kInv); b = b < 0 ? 0 : (b > 31 ? 31 : b);
                lsum += v.w; atomicAdd(&myh[b], 1u);
            }
        }
    }

    // wave32 shuffle reduction of the partial sum, one LDS f32 atomic per wave
#pragma unroll
    for (int off = 16; off > 0; off >>= 1)
        lsum += __shfl_down(lsum, off, 32);
    if (lane == 0) atomicAdd(&ssum, lsum);                 // ds_add_f32
    __syncthreads();

    // ---- pass 2: wave 0 merges 32 sub-histograms; lane==bin; first-max argmax ----
    if (wave == 0) {
        unsigned cnt = 0;
#pragma unroll
        for (int w = 0; w < NWAVES; ++w) cnt += whist[w * NBINS + lane];
        int b = lane;
#pragma unroll
        for (int off = 16; off > 0; off >>= 1) {
            unsigned oc = __shfl_xor(cnt, off, 32);
            int      ob = __shfl_xor(b,   off, 32);
            if (oc > cnt || (oc == cnt && ob < b)) { cnt = oc; b = ob; }
        }
        if (lane == 0) {
            sdom    = b;
            spooled = ssum * (1.0f / 65536.0f);            // exact pow2 divide == mean
        }
    }
    __syncthreads();

    // ---- pass 3: out = x * pooled where floor(x/8) != dominant, else 0 ----
    const int   dm = sdom;
    const float p  = spooled;
    float4* o4 = reinterpret_cast<float4*>(out) + (size_t)chan * (HW / 4);

#pragma unroll
    for (int i = 0; i < (HW / 4) / TPB; ++i) {
        float4 v = t4[i * TPB + tid];
        float4 r;
        r.x = ((int)floorf(v.x * 0.125f) != dm) ? v.x * p : 0.0f;
        r.y = ((int)floorf(v.y * 0.125f) != dm) ? v.y * p : 0.0f;
        r.z = ((int)floorf(v.z * 0.125f) != dm) ? v.z * p : 0.0f;
        r.w = ((int)floorf(v.w * 0.125f) != dm) ? v.w * p : 0.0f;
        o4[i * TPB + tid] = r;                             // global_store_b128, coalesced
    }
}

extern "C" void kernel_launch(void* const* d_in, const int* in_sizes, int n_in,
                              void* d_out, int out_size, void* d_ws, size_t ws_size,
                              hipStream_t stream)
{
    const float* x   = (const float*)d_in[0];
    float*       out = (float*)d_out;

    const int nchan = in_sizes[0] / HW;                    // 16*64 = 1024 channels

    // Allow 256KB dynamic LDS (CDNA5 WGP has 320KB). Not a stream op; capture-safe.
    (void)hipFuncSetAttribute((const void*)colormoc_fused_kernel,
                              hipFuncAttributeMaxDynamicSharedMemorySize, TILE_BYTES);

    colormoc_fused_kernel<<<dim3(nchan), dim3(TPB), TILE_BYTES, stream>>>(x, out);
}